// ProbabilityDropout_37761352466766
// MI455X (gfx1250) — compile-verified
//
#include <hip/hip_runtime.h>

// ---------------------------------------------------------------------------
// ProbabilityDropout for MI455X (gfx1250, wave32)
//
// Per row r (4096 rows):
//   z[j]   = z_mean[r,d] + exp(0.5*z_var[r,d]) * epsilon[4r+k, d]   (j=k*1024+d)
//   hist   = 16384-bin fixed-width histogram of the 4096 z values
//   probs  = softmax(hist); probs < 1e-8 -> 0
//   out[r] = x[r] * probs / 0.9
//
// Bandwidth-bound (~610 MB traffic -> ~26us @ 23.3 TB/s). CDNA5 paths used:
//   * global_load_async_to_lds_b128 (ASYNCcnt) prefetch of the 64KB x row,
//     overlapped with all compute phases, drained with s_wait_asynccnt.
//   * ds_add_u32 LDS atomics for the 64KB per-row histogram (320KB WGP LDS).
//   * wave32 shuffle reductions + small LDS tree across 32 waves.
//   * 128-bit LDS reads + float4 nontemporal global stores on the hot path.
// ---------------------------------------------------------------------------

#define BATCH     4096
#define DIM       1024
#define NOUT      16384
#define MULT      4
#define NTHREADS  1024
#define NWAVES    (NTHREADS / 32)

#define SCALE_FACTOR 0.9f
#define ZERO_POINT   1e-8f

// LDS layout (dynamic shared):
//   [0      , 65536 )  unsigned counts[16384]
//   [65536  , 131072)  float    xtile[16384]
//   [131072 , +272  )  float    red[2*NWAVES], float bc[4]
#define SMEM_BYTES (65536 + 65536 + (2 * NWAVES) * 4 + 4 * 4)

// Exact pointee type of the async builtins: int __attribute__((vector_size(16)))
typedef int v4i __attribute__((vector_size(16)));
typedef __attribute__((address_space(1))) v4i v4i_g;   // global (AS1)
typedef __attribute__((address_space(3))) v4i v4i_l;   // LDS (AS3)

typedef unsigned v4u __attribute__((ext_vector_type(4)));
typedef float    v4f __attribute__((ext_vector_type(4)));

#if __has_builtin(__builtin_amdgcn_global_load_async_to_lds_b128)
#define HAVE_ASYNC_BUILTIN 1
#endif

__device__ __forceinline__ void async_load_b128(const float* g, float* l) {
#ifdef HAVE_ASYNC_BUILTIN
  __builtin_amdgcn_global_load_async_to_lds_b128(
      (v4i_g*)(unsigned long long)g,              // AS1 src (const stripped via int)
      (v4i_l*)(unsigned)(unsigned long long)l,    // AS3 dst (low 32 bits = LDS offset)
      /*offset=*/0, /*cpol=*/0);
#else
  unsigned loff = (unsigned)(unsigned long long)(void*)l;
  asm volatile("global_load_async_to_lds_b128 %0, %1, off"
               :
               : "v"(loff), "v"(g)
               : "memory");
#endif
}

__device__ __forceinline__ void wait_async0() {
#if __has_builtin(__builtin_amdgcn_s_wait_asynccnt)
  __builtin_amdgcn_s_wait_asynccnt(0);
#else
  asm volatile("s_wait_asynccnt 0" ::: "memory");
#endif
}

__device__ __forceinline__ float wave_min(float v) {
  for (int o = 16; o > 0; o >>= 1) v = fminf(v, __shfl_xor(v, o, 32));
  return v;
}
__device__ __forceinline__ float wave_max(float v) {
  for (int o = 16; o > 0; o >>= 1) v = fmaxf(v, __shfl_xor(v, o, 32));
  return v;
}
__device__ __forceinline__ float wave_sum(float v) {
  for (int o = 16; o > 0; o >>= 1) v += __shfl_xor(v, o, 32);
  return v;
}

__global__ __launch_bounds__(NTHREADS, 1)
void ProbabilityDropout_37761352466766_kernel(
    const float* __restrict__ z_mean,   // [4096, 1024]
    const float* __restrict__ z_var,    // [4096, 1024]
    const float* __restrict__ x,        // [4096, 16384]
    const float* __restrict__ epsilon,  // [16384, 1024]
    float* __restrict__ out)            // [4096, 16384]
{
  extern __shared__ char smem[];
  unsigned* counts = (unsigned*)smem;                  // 16384 bins
  float*    xtile  = (float*)(smem + 65536);           // 16384 floats
  float*    red    = (float*)(smem + 131072);          // 2*NWAVES floats
  float*    bc     = red + 2 * NWAVES;                 // 4 floats

  v4u* counts4 = (v4u*)counts;                         // 4096 x uint4
  v4f* xtile4  = (v4f*)xtile;                          // 4096 x float4

  const int t    = threadIdx.x;
  const int r    = blockIdx.x;
  const int lane = t & 31;
  const int wid  = t >> 5;

  // ---- kick off async DMA of the x row into LDS (overlaps everything) ----
  const float* xrow = x + (size_t)r * NOUT;
  #pragma unroll
  for (int k = 0; k < 4; ++k) {
    int e = k * 4096 + t * 4;                 // 16B / lane / issue, fully coalesced
    async_load_b128(xrow + e, xtile + e);
  }

  // ---- zero histogram bins (uint4 LDS stores) ----
  #pragma unroll
  for (int k = 0; k < 4; ++k) counts4[t + k * NTHREADS] = (v4u)(0u);

  // ---- reparameterization: 4 z-values per thread, kept in registers ----
  const float zm = __builtin_nontemporal_load(&z_mean[(size_t)r * DIM + t]);
  const float zv = __builtin_nontemporal_load(&z_var[(size_t)r * DIM + t]);
  const float sv = expf(0.5f * zv);
  float zl[MULT];
  float vmin = INFINITY, vmax = -INFINITY;
  #pragma unroll
  for (int k = 0; k < MULT; ++k) {
    float e = __builtin_nontemporal_load(&epsilon[((size_t)(r * MULT + k)) * DIM + t]);
    float z = fmaf(sv, e, zm);
    zl[k] = z;
    vmin = fminf(vmin, z);
    vmax = fmaxf(vmax, z);
  }

  // ---- block min/max reduction (wave32 shuffles + 32-wave LDS tree) ----
  vmin = wave_min(vmin);
  vmax = wave_max(vmax);
  if (lane == 0) { red[wid] = vmin; red[NWAVES + wid] = vmax; }
  __syncthreads();
  if (wid == 0) {
    float m0 = wave_min(red[lane]);
    float m1 = wave_max(red[NWAVES + lane]);
    if (lane == 0) { bc[0] = m0; bc[1] = m1; }
  }
  __syncthreads();
  const float mn  = bc[0];
  const float mx  = bc[1];
  const float inv = 1.0f / (mx - mn);

  // ---- histogram via LDS atomics (ds_add_u32) ----
  #pragma unroll
  for (int k = 0; k < MULT; ++k) {
    float scaled = (zl[k] - mn) * inv;
    int idx = (int)floorf(scaled * (float)NOUT);
    idx = min(max(idx, 0), NOUT - 1);
    atomicAdd(&counts[idx], 1u);
  }
  __syncthreads();

  // ---- softmax over 16384 counts: max, then sum of exp (b128 LDS reads) ----
  v4f cf[4];
  float cmax = 0.0f;
  #pragma unroll
  for (int k = 0; k < 4; ++k) {
    v4u c = counts4[t + k * NTHREADS];
    #pragma unroll
    for (int c4 = 0; c4 < 4; ++c4) {
      float f = (float)c[c4];
      cf[k][c4] = f;
      cmax = fmaxf(cmax, f);
    }
  }
  cmax = wave_max(cmax);
  if (lane == 0) red[wid] = cmax;
  __syncthreads();
  if (wid == 0) {
    float m = wave_max(red[lane]);
    if (lane == 0) bc[2] = m;
  }
  __syncthreads();
  const float maxc = bc[2];

  v4f pl[4];
  float s = 0.0f;
  #pragma unroll
  for (int k = 0; k < 4; ++k) {
    #pragma unroll
    for (int c4 = 0; c4 < 4; ++c4) {
      float p = expf(cf[k][c4] - maxc);
      pl[k][c4] = p;
      s += p;
    }
  }
  s = wave_sum(s);
  if (lane == 0) red[wid] = s;
  __syncthreads();
  if (wid == 0) {
    float m = wave_sum(red[lane]);
    if (lane == 0) bc[3] = m;
  }
  __syncthreads();
  const float invsum = 1.0f / bc[3];

  // ---- drain async x tile, then fused threshold + scale + NT store ----
  wait_async0();
  __syncthreads();

  const float invscale = 1.0f / SCALE_FACTOR;
  v4f* orow4 = (v4f*)(out + (size_t)r * NOUT);
  #pragma unroll
  for (int k = 0; k < 4; ++k) {
    v4f xv = xtile4[t + k * NTHREADS];
    v4f o;
    #pragma unroll
    for (int c4 = 0; c4 < 4; ++c4) {
      float p = pl[k][c4] * invsum;
      p = (p < ZERO_POINT) ? 0.0f : p;
      o[c4] = xv[c4] * p * invscale;
    }
    __builtin_nontemporal_store(o, &orow4[t + k * NTHREADS]);
  }
}

extern "C" void kernel_launch(void* const* d_in, const int* in_sizes, int n_in,
                              void* d_out, int out_size, void* d_ws, size_t ws_size,
                              hipStream_t stream) {
  (void)in_sizes; (void)n_in; (void)out_size; (void)d_ws; (void)ws_size;
  const float* z_mean  = (const float*)d_in[0];
  const float* z_var   = (const float*)d_in[1];
  const float* x       = (const float*)d_in[2];
  const float* epsilon = (const float*)d_in[3];
  float* out = (float*)d_out;

  // 128.3 KB dynamic LDS per workgroup (gfx1250 WGP has 320 KB).
  (void)hipFuncSetAttribute(
      reinterpret_cast<const void*>(ProbabilityDropout_37761352466766_kernel),
      hipFuncAttributeMaxDynamicSharedMemorySize, SMEM_BYTES);

  ProbabilityDropout_37761352466766_kernel<<<dim3(BATCH), dim3(NTHREADS),
                                             SMEM_BYTES, stream>>>(
      z_mean, z_var, x, epsilon, out);
}